// inter_conv_net_3178275799092
// MI455X (gfx1250) — compile-verified
//
#include <hip/hip_runtime.h>

typedef float v2f __attribute__((ext_vector_type(2)));
typedef float v4f __attribute__((ext_vector_type(4)));
typedef float v8f __attribute__((ext_vector_type(8)));

// D = A(16x4, f32) * B(4x16, f32) + C(16x16, f32)
#define WMMA_F32(A, B, C) \
  __builtin_amdgcn_wmma_f32_16x16x4_f32(false, (A), false, (B), (short)0, (C), false, false)

// Single-instruction relu: VOP2 v_max_num_f32 with inline-constant 0 in src0.
// (WMMA output is never sNaN, so IEEE maxNum without canonicalize is exact.)
static __device__ __forceinline__ float relu1(float v) {
  float r;
  asm("v_max_num_f32 %0, 0, %1" : "=v"(r) : "v"(v));
  return r;
}
static __device__ __forceinline__ float maxf1(float a, float b) {
  float r;
  asm("v_max_num_f32 %0, %1, %2" : "=v"(r) : "v"(a), "v"(b));
  return r;
}

// DPP lane-permute within a 16-lane row (pure VALU, no LDS traffic).
template <int CTRL>
static __device__ __forceinline__ float dpp_mov(float v) {
  int r = __builtin_amdgcn_update_dpp(0, __float_as_int(v), CTRL, 0xF, 0xF, true);
  return __int_as_float(r);
}
#define DPP_XOR1  0xB1   // quad_perm(1,0,3,2)
#define DPP_XOR2  0x4E   // quad_perm(2,3,0,1)
#define DPP_HMIR  0x141  // row_half_mirror (lane^7)
#define DPP_MIR   0x140  // row_mirror      (lane^15)

// One wave (32 lanes) owns one (batch, h) row of 256 pixels, processed as
// 4 chunks of 64 pixels; lane p owns pixels {64t+4p .. 64t+4p+3} so each
// channel is one coalesced global_load_b128 feeding 4 sub-tiles.
//
// Inter-layer reshape is FREE: the K-dimension of each GEMM is permuted
// (sigma(4kb+j)=2kb+j, sigma(4kb+2+j)=8+2kb+j) so that the B fragment of
// layer n+1 is exactly the adjacent register pairs of layer n's relu'd
// accumulator -- no cross-lane traffic at all. Only the weights' A fragments
// are loaded with permuted columns (loop-invariant).
__global__ __launch_bounds__(256) void inter_conv_net_rows(
    const float* __restrict__ x,   // [B,5,256,256]
    const float* __restrict__ W0, const float* __restrict__ b0,   // [16,5],[16]
    const float* __restrict__ W1, const float* __restrict__ b1,   // [16,16],[16]
    const float* __restrict__ W2, const float* __restrict__ b2,   // [6,16],[6]
    float* __restrict__ out,       // [B,12,256]
    int nrows)                     // B*256
{
  const int lane = threadIdx.x & 31;
  const int wave = threadIdx.x >> 5;
  const int row  = blockIdx.x * 8 + wave;
  if (row >= nrows) return;                 // whole-wave uniform exit
  const int bidx = row >> 8;
  const int h    = row & 255;

  const bool lo = lane < 16;                // lower half-wave
  const int  hb = lo ? 0 : 1;
  const int  p  = lane & 15;                // lane slot / A-matrix row
  const int  m  = p;

  // ---------------- loop-invariant A fragments (weights) ----------------
  // Layer 1: W0 [16,5] (stride 5); K padded 5 -> 8 => two K-blocks.
  v2f a1_0, a1_1;
  a1_0.x = W0[m * 5 + (lo ? 0 : 2)];
  a1_0.y = W0[m * 5 + (lo ? 1 : 3)];
  a1_1.x = lo ? W0[m * 5 + 4] : 0.0f;       // K=4 real, K=5..7 zero
  a1_1.y = 0.0f;

  // Layers 2/3: K-permutation sigma => this half's column pair is 2kb + 8*hb.
  v2f a2[4], a3[4];
#pragma unroll
  for (int kb = 0; kb < 4; ++kb) {
    const int c0 = 2 * kb + 8 * hb;
    a2[kb].x = W1[m * 16 + c0];
    a2[kb].y = W1[m * 16 + c0 + 1];
    a3[kb].x = (m < 6) ? W2[m * 16 + c0]     : 0.0f;   // pad out-ch 6..15
    a3[kb].y = (m < 6) ? W2[m * 16 + c0 + 1] : 0.0f;
  }

  // ---------------- biases pre-placed in C accumulator layout ----------------
  // C layout: VGPR r -> out-ch r (lanes 0-15) / r+8 (lanes 16-31).
  v8f cb0, cb1, cb2;
#pragma unroll
  for (int r = 0; r < 8; ++r) {
    cb0[r] = b0[r + 8 * hb];
    cb1[r] = b1[r + 8 * hb];
    cb2[r] = (!hb && r < 6) ? b2[r] : 0.0f; // padded out-chs: 0 bias -> relu(0)=0
  }

  const long cs    = 256L * 256L;                        // channel stride
  const long xbase = ((long)bidx * 5 * 256 + h) * 256    // &x[b,0,h,0]
                   + 4 * p;                              // this lane's pixel group
  const long cA = (long)(lo ? 0 : 2) * cs;
  const long cB = (long)(lo ? 1 : 3) * cs;
  const long cC = 4 * cs;

  float sum[6], mx[6];
#pragma unroll
  for (int r = 0; r < 6; ++r) { sum[r] = 0.0f; mx[r] = 0.0f; }  // relu >= 0

  // preload chunk 0 (3 x global_load_b128 per lane)
  v4f qA = *(const v4f*)(x + xbase + cA);
  v4f qB = *(const v4f*)(x + xbase + cB);
  v4f qC = *(const v4f*)(x + xbase + cC);

#pragma unroll
  for (int t = 0; t < 4; ++t) {
    v4f nA, nB, nC;
    if (t < 3) {                            // software-pipeline next chunk
      const long nb = xbase + 64 * (t + 1);
      nA = *(const v4f*)(x + nb + cA);
      nB = *(const v4f*)(x + nb + cB);
      nC = *(const v4f*)(x + nb + cC);
    }

#pragma unroll
    for (int i = 0; i < 4; ++i) {           // 4 sub-tiles of 16 pixels
      // ---- layer 1: B tile = x[ch, pixel], K padded to 8 ----
      v2f bx0, bx1;
      bx0.x = qA[i];
      bx0.y = qB[i];
      bx1.x = lo ? qC[i] : 0.0f;
      bx1.y = 0.0f;

      v8f acc = cb0;
      acc = WMMA_F32(a1_0, bx0, acc);
      acc = WMMA_F32(a1_1, bx1, acc);

      // relu; thanks to the K-permutation, adjacent pairs of hv ARE the
      // next layer's B fragments (same registers in both half-waves).
      v8f hv;
#pragma unroll
      for (int r = 0; r < 8; ++r) hv[r] = relu1(acc[r]);

      // ---- layer 2: 16 -> 16, K = 16 => 4 chained WMMAs ----
      acc = cb1;
#pragma unroll
      for (int kb = 0; kb < 4; ++kb) {
        v2f bb; bb.x = hv[2 * kb]; bb.y = hv[2 * kb + 1];
        acc = WMMA_F32(a2[kb], bb, acc);
      }

      v8f gv;
#pragma unroll
      for (int r = 0; r < 8; ++r) gv[r] = relu1(acc[r]);

      // ---- layer 3: 16 -> 6 (padded to 16), K = 16 => 4 chained WMMAs ----
      acc = cb2;
#pragma unroll
      for (int kb = 0; kb < 4; ++kb) {
        v2f bc; bc.x = gv[2 * kb]; bc.y = gv[2 * kb + 1];
        acc = WMMA_F32(a3[kb], bc, acc);
      }

      // relu + running sum/max (upper half holds padded chs = 0)
#pragma unroll
      for (int r = 0; r < 6; ++r) {
        const float v = relu1(acc[r]);
        sum[r] += v;
        mx[r]   = maxf1(mx[r], v);
      }
    }

    qA = nA; qB = nB; qC = nC;
  }

  // ---- reduction over the 16 lane slots: pure-VALU DPP butterfly ----
  // Real partials live in lanes 0-15 only (upper half = padded zeros), and
  // only lane 0 stores, so 4 steps within the 16-lane DPP row suffice.
#pragma unroll
  for (int r = 0; r < 6; ++r) {
    sum[r] += dpp_mov<DPP_XOR1>(sum[r]);
    mx[r]   = maxf1(mx[r], dpp_mov<DPP_XOR1>(mx[r]));
    sum[r] += dpp_mov<DPP_XOR2>(sum[r]);
    mx[r]   = maxf1(mx[r], dpp_mov<DPP_XOR2>(mx[r]));
    sum[r] += dpp_mov<DPP_HMIR>(sum[r]);
    mx[r]   = maxf1(mx[r], dpp_mov<DPP_HMIR>(mx[r]));
    sum[r] += dpp_mov<DPP_MIR>(sum[r]);
    mx[r]   = maxf1(mx[r], dpp_mov<DPP_MIR>(mx[r]));
  }

  if (lane == 0) {
    const long ob = (long)bidx * 12 * 256 + h;
#pragma unroll
    for (int r = 0; r < 6; ++r) {
      out[ob + (long)r * 256]       = sum[r];   // fea_sum -> channels 0..5
      out[ob + (long)(6 + r) * 256] = mx[r];    // fea_max -> channels 6..11
    }
  }
}

extern "C" void kernel_launch(void* const* d_in, const int* in_sizes, int n_in,
                              void* d_out, int out_size, void* d_ws, size_t ws_size,
                              hipStream_t stream) {
  (void)n_in; (void)d_ws; (void)ws_size; (void)out_size;
  const float* x  = (const float*)d_in[0];
  const float* W0 = (const float*)d_in[1];
  const float* b0 = (const float*)d_in[2];
  const float* W1 = (const float*)d_in[3];
  const float* b1 = (const float*)d_in[4];
  const float* W2 = (const float*)d_in[5];
  const float* b2 = (const float*)d_in[6];
  float* out = (float*)d_out;

  const int B      = in_sizes[0] / (5 * 256 * 256);  // 64 for the reference shapes
  const int nrows  = B * 256;                        // one wave per (b,h) row
  const int blocks = (nrows + 7) / 8;                // 8 waves (256 threads) per block

  inter_conv_net_rows<<<blocks, 256, 0, stream>>>(x, W0, b0, W1, b1, W2, b2, out, nrows);
}